// CapsuleLayer_36369783062608
// MI455X (gfx1250) — compile-verified
//
#include <hip/hip_runtime.h>

// Capsule routing, fully fused: u_hat never touches HBM.
// v_wmma_f32_16x16x4_f32 (native CDNA5 fp32 matrix op), M=batch tile,
// N = 16 j's of one output capsule o, K=d (8 = 2 chained WMMAs).
// Cross-lane j-reductions done with DPP16 butterflies (pure VALU, no LDS pipe).

typedef __attribute__((ext_vector_type(2))) float v2f;
typedef __attribute__((ext_vector_type(8))) float v8f;

#define B_TOT   256
#define I_TOT   1152
#define O_CAPS  10
#define J_DIM   16
#define D_DIM   8
#define NBB     (B_TOT / 16)          // 16 batch tiles
#define ISPLIT  16                    // i-dimension split across blocks
#define I_PER_BLK (I_TOT / ISPLIT)    // 72
#define WAVES   4
#define I_PER_WAVE (I_PER_BLK / WAVES) // 18
#define SOJ     (O_CAPS * J_DIM)      // 160
#define S_ELEMS (B_TOT * SOJ)         // 40960 floats

// DPP16 control encodings
#define DPP_QUAD_XOR1   0xB1   // quad_perm(1,0,3,2)
#define DPP_QUAD_XOR2   0x4E   // quad_perm(2,3,0,1)
#define DPP_ROW_HMIRROR 0x141  // row_half_mirror
#define DPP_ROW_MIRROR  0x140  // row_mirror

template <int CTRL>
__device__ __forceinline__ float dpp_add(float v) {
    int p = __builtin_amdgcn_mov_dpp(__float_as_int(v), CTRL, 0xf, 0xf, true);
    return v + __int_as_float(p);
}

__device__ __forceinline__ float reduce16(float v) {
    // butterfly sum across each 16-lane half; result replicated to all 16 lanes
    v = dpp_add<DPP_QUAD_XOR1>(v);
    v = dpp_add<DPP_QUAD_XOR2>(v);
    v = dpp_add<DPP_ROW_HMIRROR>(v);
    v = dpp_add<DPP_ROW_MIRROR>(v);
    return v;
}

__global__ void caps_zero(float* __restrict__ p, int n) {
    int t = blockIdx.x * blockDim.x + threadIdx.x;
    if (t < n) p[t] = 0.0f;
}

// One routing iteration's partial-sum pass.
// block = (batch tile bb, i-split is); 4 waves; each wave owns 18 i's and all 10 o's.
__global__ __launch_bounds__(128, 1)
void caps_partial(const float* __restrict__ x,     // [256][1152][8]
                  const float* __restrict__ W,     // [1152][10][16][8]
                  const float* __restrict__ Vacc,  // [256][10][16]  (sum of previous v's)
                  float* __restrict__ s_glob)      // [256][10][16]  (zeroed before launch)
{
    __shared__ float s_blk[16 * SOJ];              // 10 KB block accumulator

    const int bb    = blockIdx.x & (NBB - 1);
    const int is    = blockIdx.x / NBB;
    const int wave  = threadIdx.x >> 5;
    const int lane  = threadIdx.x & 31;
    const int half  = lane >> 4;                   // fragment half (M +0 / +8)
    const int j     = lane & 15;                   // N index within o-tile = j
    const int brow0 = bb * 16 + 8 * half;          // + r = global batch row of C/D frag

    for (int t = threadIdx.x; t < 16 * SOJ; t += 128) s_blk[t] = 0.0f;
    __syncthreads();

    // Vacc fragment, laid out exactly like the WMMA C/D fragment per o
    float vfrag[O_CAPS][8];
    #pragma unroll
    for (int o = 0; o < O_CAPS; ++o)
        #pragma unroll
        for (int r = 0; r < 8; ++r)
            vfrag[o][r] = Vacc[(size_t)(brow0 + r) * SOJ + o * 16 + j];

    float sfrag[O_CAPS][8];
    #pragma unroll
    for (int o = 0; o < O_CAPS; ++o)
        #pragma unroll
        for (int r = 0; r < 8; ++r) sfrag[o][r] = 0.0f;

    const int i_base = is * I_PER_BLK;
    const int k0 = 2 * half;                       // K slot per lane-half (A & B layouts)

    for (int t = 0; t < I_PER_WAVE; ++t) {
        const int i = i_base + wave + WAVES * t;

        // A fragment: A[M=16 batches][K=8 d's]; M = lane&15, K split across halves/VGPRs
        const float* xp = x + (size_t)(bb * 16 + j) * (I_TOT * D_DIM) + (size_t)i * D_DIM;
        v2f a_lo = { xp[k0],     xp[k0 + 1] };     // K = {0,1} / {2,3}
        v2f a_hi = { xp[k0 + 4], xp[k0 + 5] };     // K = {4,5} / {6,7}

        if (t + 1 < I_PER_WAVE)
            __builtin_prefetch(W + (size_t)(i + WAVES) * (O_CAPS * J_DIM * D_DIM), 0, 1);

        // u_hat tiles for all 10 capsules, kept in registers
        float dts[O_CAPS][8];
        #pragma unroll
        for (int o = 0; o < O_CAPS; ++o) {
            // B fragment: B[K=4][N=16 j's]; W is [i][o][j][d], d innermost
            const float* wp = W + (size_t)i * (O_CAPS * J_DIM * D_DIM)
                                + o * (J_DIM * D_DIM) + j * D_DIM;
            v2f b_lo = { wp[k0],     wp[k0 + 1] };
            v2f b_hi = { wp[k0 + 4], wp[k0 + 5] };
            v8f d = {};
            d = __builtin_amdgcn_wmma_f32_16x16x4_f32(false, a_lo, false, b_lo,
                                                      (short)0, d, false, false);
            d = __builtin_amdgcn_wmma_f32_16x16x4_f32(false, a_hi, false, b_hi,
                                                      (short)0, d, false, false);
            #pragma unroll
            for (int r = 0; r < 8; ++r) dts[o][r] = d[r];
        }

        // Routing: logits l[b,o] = sum_j u_hat*Vacc, softmax over o, accumulate s.
        // Per r, both 16-lane halves end holding identical l -> elementwise c*u works.
        #pragma unroll
        for (int r = 0; r < 8; ++r) {
            float l[O_CAPS], e[O_CAPS];
            #pragma unroll
            for (int o = 0; o < O_CAPS; ++o)
                l[o] = reduce16(dts[o][r] * vfrag[o][r]);
            float m = l[0];
            #pragma unroll
            for (int o = 1; o < O_CAPS; ++o) m = fmaxf(m, l[o]);
            float z = 0.0f;
            #pragma unroll
            for (int o = 0; o < O_CAPS; ++o) { e[o] = __expf(l[o] - m); z += e[o]; }
            const float rz = 1.0f / z;
            #pragma unroll
            for (int o = 0; o < O_CAPS; ++o)
                sfrag[o][r] = fmaf(e[o] * rz, dts[o][r], sfrag[o][r]);
        }
    }

    // combine 4 waves in LDS (ds_add_f32), then one global atomic per element
    #pragma unroll
    for (int o = 0; o < O_CAPS; ++o)
        #pragma unroll
        for (int r = 0; r < 8; ++r)
            atomicAdd(&s_blk[(r + 8 * half) * SOJ + o * 16 + j], sfrag[o][r]);
    __syncthreads();

    for (int t = threadIdx.x; t < 16 * SOJ; t += 128)
        atomicAdd(&s_glob[(size_t)(bb * 16 + t / SOJ) * SOJ + (t % SOJ)], s_blk[t]);
}

// squash(s) -> v ; Vacc += v ; optionally emit v. One (b,o) pair per 16-lane half.
__global__ __launch_bounds__(128, 1)
void caps_squash(const float* __restrict__ s, float* __restrict__ Vacc,
                 float* __restrict__ out, int write_out)
{
    const int lane = threadIdx.x & 31;
    const int wave = threadIdx.x >> 5;
    const int pair = blockIdx.x * 8 + wave * 2 + (lane >> 4);  // 0..2559 = b*10+o
    const int j    = lane & 15;

    const float sv  = s[pair * 16 + j];
    const float n2  = reduce16(sv * sv);
    const float nrm = sqrtf(n2);
    const float sc  = (n2 / (1.0f + n2)) / (nrm + 1e-8f);
    const float v   = sc * sv;

    Vacc[pair * 16 + j] += v;
    if (write_out) out[pair * 16 + j] = v;
}

extern "C" void kernel_launch(void* const* d_in, const int* in_sizes, int n_in,
                              void* d_out, int out_size, void* d_ws, size_t ws_size,
                              hipStream_t stream) {
    (void)in_sizes; (void)n_in; (void)out_size; (void)ws_size;
    const float* x = (const float*)d_in[0];   // [256,1152,8]
    const float* W = (const float*)d_in[1];   // [1152,10,16,8]
    float* out  = (float*)d_out;              // [256,10,16]
    float* Vacc = (float*)d_ws;               // 40960 floats
    float* s    = Vacc + S_ELEMS;             // 40960 floats (ws need: 320 KB)

    caps_zero<<<(S_ELEMS + 255) / 256, 256, 0, stream>>>(Vacc, S_ELEMS);
    for (int it = 0; it < 3; ++it) {
        caps_zero<<<(S_ELEMS + 255) / 256, 256, 0, stream>>>(s, S_ELEMS);
        caps_partial<<<NBB * ISPLIT, 128, 0, stream>>>(x, W, Vacc, s);
        caps_squash<<<(B_TOT * O_CAPS) / 8, 128, 0, stream>>>(s, Vacc, out, it == 2 ? 1 : 0);
    }
}